// TurboQuantRotation_19576460935345
// MI455X (gfx1250) — compile-verified
//
#include <hip/hip_runtime.h>

// Cascaded SRHT on MI455X (gfx1250, wave32).
// H_1024 = H16 (x) H4 (x) H16  -> two WMMA_F32_16X16X4_F32 contraction stages
// + one register-level H4 butterfly per pass. One row (1024 f32) per wave,
// wave-private LDS double buffer, no barriers needed (LDS ops are in-order
// within a wave). HBM-bound: 512 MB total traffic ~= 22 us at 23.3 TB/s.

typedef float v2f __attribute__((ext_vector_type(2)));
typedef float v8f __attribute__((ext_vector_type(8)));

#define WAVES_PER_BLOCK 8
#define ROW_LEN 1024

// Compute D(16x16) = X[tile*16 .. tile*16+15, 0..15] @ B(16x16),
// where X lives in LDS row-major [64 x 16] and B is given as four
// 4x16 K-chunk fragments (resident in registers).
// f32 WMMA layouts (ISA 7.12.2):
//   A 16x4 : lane m (and m+16) holds M=m; VGPR0 = K {0 | 2}, VGPR1 = K {1 | 3}
//   B 4x16 : lane n (and n+16) holds N=n; VGPR0 = K {0 | 2}, VGPR1 = K {1 | 3}
//   C/D    : VGPR v holds M = v (lanes 0-15) / v+8 (lanes 16-31), N = lane&15
__device__ __forceinline__ v8f wmma_rowtile(const float* lds_mat, int tile,
                                            const v2f hfrag[4]) {
  const int lane = threadIdx.x & 31;
  const int m    = lane & 15;
  const int koff = (lane >> 4) ? 2 : 0;
  v8f c = {};
#pragma unroll
  for (int kb = 0; kb < 4; ++kb) {
    const float2 av =
        *reinterpret_cast<const float2*>(lds_mat + (tile * 16 + m) * 16 + kb * 4 + koff);
    v2f a;
    a.x = av.x;  // K = kb*4 + koff
    a.y = av.y;  // K = kb*4 + koff + 1
    c = __builtin_amdgcn_wmma_f32_16x16x4_f32(
        /*neg_a=*/false, a, /*neg_b=*/false, hfrag[kb],
        /*c_mod=*/(short)0, c, /*reuse_a=*/false, /*reuse_b=*/false);
  }
  return c;
}

__global__ void __launch_bounds__(256) srht_wmma_kernel(
    const float* __restrict__ x, const float* __restrict__ signs,
    float* __restrict__ out, int rows) {
  __shared__ float buf1[WAVES_PER_BLOCK][ROW_LEN];  // natural-order row
  __shared__ float buf2[WAVES_PER_BLOCK][ROW_LEN];  // (f,g)-transposed view

  const int wave = threadIdx.x >> 5;
  const int lane = threadIdx.x & 31;
  const int row  = blockIdx.x * WAVES_PER_BLOCK + wave;
  if (row >= rows) return;  // wave-uniform: EXEC stays all-1s for live waves

  float* b1 = buf1[wave];
  float* b2 = buf2[wave];

  // ---- B fragments for H16: H16[k][n] = (-1)^popcount(k & n) -------------
  v2f hfrag[4];
  {
    const int n    = lane & 15;
    const int koff = (lane >> 4) ? 2 : 0;
#pragma unroll
    for (int kb = 0; kb < 4; ++kb) {
      const int k0 = kb * 4 + koff;
      hfrag[kb].x = (__popc(k0 & n) & 1) ? -1.0f : 1.0f;
      hfrag[kb].y = (__popc((k0 + 1) & n) & 1) ? -1.0f : 1.0f;
    }
  }

  // ---- load row * signs[0] into LDS (coalesced b128) ---------------------
  const float4* xrow = reinterpret_cast<const float4*>(x + (size_t)row * ROW_LEN);
  const float4* s0v  = reinterpret_cast<const float4*>(signs);
  const float4* s1v  = reinterpret_cast<const float4*>(signs + ROW_LEN);
  float4*       b1v  = reinterpret_cast<float4*>(b1);
#pragma unroll
  for (int c = 0; c < 8; ++c) {
    const int i4 = c * 32 + lane;  // 256 float4 per row
    const float4 xv = xrow[i4];
    const float4 sv = s0v[i4];
    b1v[i4] = make_float4(xv.x * sv.x, xv.y * sv.y, xv.z * sv.z, xv.w * sv.w);
  }

  const float scale = 0.03125f;  // 1/sqrt(1024)

#pragma unroll
  for (int pass = 0; pass < 2; ++pass) {
    // ---- Stage 1: contract b (fast index). X1[a,b] = b1[a*16+b], a=f*4+g.
    //      Y1 = X1 @ H16; scatter to b2[(g*16 + b')*16 + f].
    const int col = lane & 15;
    const int hi8 = (lane >> 4) ? 8 : 0;
#pragma unroll
    for (int t = 0; t < 4; ++t) {
      const v8f d = wmma_rowtile(b1, t, hfrag);
#pragma unroll
      for (int v = 0; v < 8; ++v) {
        const int a = t * 16 + v + hi8;  // row index of Y1
        const int f = a >> 2;
        const int g = a & 3;
        b2[(g * 16 + col) * 16 + f] = d[v];
      }
    }

    // ---- Stage 2: contract f. X2[g*16+b', f] = b2[...]; tile index == g.
    v8f T[4];
#pragma unroll
    for (int t = 0; t < 4; ++t) T[t] = wmma_rowtile(b2, t, hfrag);

    // ---- Stage 3: H4 butterfly across g (pure VGPR ops on accumulators).
    const v8f u0 = T[0] + T[1];
    const v8f u1 = T[0] - T[1];
    const v8f u2 = T[2] + T[3];
    const v8f u3 = T[2] - T[3];
    v8f O[4];
    O[0] = u0 + u2;
    O[1] = u1 + u3;
    O[2] = u0 - u2;
    O[3] = u1 - u3;

    // O[g'] element (M=b', N=f') -> natural index q = f'*64 + g'*16 + b'.
    // The 8 accumulator rows per lane are 8 consecutive q -> two b128 stores.
    const int fp = lane & 15;
#pragma unroll
    for (int gp = 0; gp < 4; ++gp) {
      const int qb = fp * 64 + gp * 16 + hi8;
      float4* dst  = reinterpret_cast<float4*>(b1 + qb);
      const v8f o  = O[gp];
      if (pass == 0) {
        // fuse scale and signs[1] for the next pass
        const float4 sA = s1v[qb >> 2];
        const float4 sB = s1v[(qb >> 2) + 1];
        dst[0] = make_float4(o[0] * scale * sA.x, o[1] * scale * sA.y,
                             o[2] * scale * sA.z, o[3] * scale * sA.w);
        dst[1] = make_float4(o[4] * scale * sB.x, o[5] * scale * sB.y,
                             o[6] * scale * sB.z, o[7] * scale * sB.w);
      } else {
        dst[0] = make_float4(o[0] * scale, o[1] * scale, o[2] * scale, o[3] * scale);
        dst[1] = make_float4(o[4] * scale, o[5] * scale, o[6] * scale, o[7] * scale);
      }
    }
  }

  // ---- perfectly coalesced b128 store of the finished row ----------------
  float4* orow = reinterpret_cast<float4*>(out + (size_t)row * ROW_LEN);
#pragma unroll
  for (int c = 0; c < 8; ++c) {
    const int i4 = c * 32 + lane;
    orow[i4] = b1v[i4];
  }
}

extern "C" void kernel_launch(void* const* d_in, const int* in_sizes, int n_in,
                              void* d_out, int out_size, void* d_ws, size_t ws_size,
                              hipStream_t stream) {
  (void)n_in; (void)out_size; (void)d_ws; (void)ws_size;
  const float* x     = (const float*)d_in[0];
  const float* signs = (const float*)d_in[1];
  // d_in[2] = h_mat: unused — H16 fragments are synthesized from popcount parity.
  float* out = (float*)d_out;

  const int rows   = in_sizes[0] / ROW_LEN;              // 8*8192 = 65536
  const int blocks = (rows + WAVES_PER_BLOCK - 1) / WAVES_PER_BLOCK;
  hipLaunchKernelGGL(srht_wmma_kernel, dim3(blocks), dim3(256), 0, stream,
                     x, signs, out, rows);
}